// PointPillarsScatter_38534446580425
// MI455X (gfx1250) — compile-verified
//
#include <hip/hip_runtime.h>
#include <stdint.h>

// Canvas geometry (hardcoded in the reference module).
#define NX_DIM 400
#define NY_DIM 400
#define NYNX   (NY_DIM * NX_DIM)   // 160000 floats per (b,c) plane

typedef float vfloat4 __attribute__((ext_vector_type(4)));

// ---------------------------------------------------------------------------
// CDNA5 async global->LDS helpers (gfx1250). Tracked with ASYNCcnt.
// ---------------------------------------------------------------------------
__device__ __forceinline__ void async_load_b128_to_lds(uint32_t lds_addr,
                                                       const void* gaddr) {
  // GLOBAL_LOAD_ASYNC_TO_LDS_B128: per-lane LDS byte address in VDST VGPR,
  // per-lane 64-bit global address in VADDR pair, GV mode (SADDR = off).
  asm volatile("global_load_async_to_lds_b128 %0, %1, off"
               :
               : "v"(lds_addr), "v"(gaddr)
               : "memory");
}

__device__ __forceinline__ void wait_asynccnt_zero() {
  asm volatile("s_wait_asynccnt 0" ::: "memory");
}

// ---------------------------------------------------------------------------
// Pass 1: zero the occupancy map (B*NY*NX ints) with wide stores.
// Default (cache-allocating) stores: we WANT the map resident in L2 — it is
// re-read 64x (once per channel plane) by the gather pass.
// ---------------------------------------------------------------------------
__global__ void zero_map_kernel(int4* __restrict__ map4, int n4) {
  int i = blockIdx.x * blockDim.x + threadIdx.x;
  if (i < n4) map4[i] = make_int4(0, 0, 0, 0);
}

// ---------------------------------------------------------------------------
// Pass 2: scatter pillar ids (p+1) into the map. Coords are unique triples,
// so this is collision-free; 96k 4-byte writes, negligible traffic.
// coords row = (b, 0, y, x) int32 -> load as one int4.
// ---------------------------------------------------------------------------
__global__ void scatter_ids_kernel(const int4* __restrict__ coords,
                                   int* __restrict__ map, int n) {
  int p = blockIdx.x * blockDim.x + threadIdx.x;
  if (p >= n) return;
  int4 cd = coords[p];              // x=b, y=0, z=y, w=x
  map[cd.x * NYNX + cd.z * NX_DIM + cd.w] = p + 1;
}

// ---------------------------------------------------------------------------
// Pass 3: gather-formulated fill. One float4 of output per thread, written
// exactly once, fully coalesced (this is the HBM-write roofline pass).
//  - map entries staged through LDS with the CDNA5 async copy engine
//    (ASYNCcnt) so the fetch overlaps index arithmetic.
//  - output stored NON-TEMPORALLY (th:TH_STORE_NT): the 327.7 MB write-once
//    stream must not sweep the L2-resident map (5.1 MB, read 64x) and
//    feature array (24.6 MB, lines shared by up to 32 channel planes).
// ---------------------------------------------------------------------------
__global__ void gather_out_kernel(const float* __restrict__ vf,
                                  const int*   __restrict__ map,
                                  float*       __restrict__ out,
                                  int nfloat4, int C) {
  __shared__ int4 smem[256];

  int t = blockIdx.x * blockDim.x + threadIdx.x;
  if (t >= nfloat4) return;

  int o     = t * 4;                // flat output index (fits in int: 81.92M)
  int per_b = C * NYNX;             // floats per batch image
  int b     = o / per_b;
  int rem   = o - b * per_b;
  int c     = rem / NYNX;
  int s     = rem - c * NYNX;       // s % 4 == 0, never crosses plane edge

  // Kick off the async map fetch for this thread's 4 spatial cells; the
  // remaining address math overlaps with the copy in flight.
  const int* gsrc = map + b * NYNX + s;
  uint32_t lds_addr = (uint32_t)(uintptr_t)(&smem[threadIdx.x]);
  async_load_b128_to_lds(lds_addr, (const void*)gsrc);

  const float* vfc = vf + c;        // channel column base

  wait_asynccnt_zero();             // our wave's async loads are complete
  int4 m = smem[threadIdx.x];       // own slot only -> no barrier needed

  vfloat4 r;
  r.x = m.x ? vfc[(m.x - 1) * C] : 0.0f;
  r.y = m.y ? vfc[(m.y - 1) * C] : 0.0f;
  r.z = m.z ? vfc[(m.z - 1) * C] : 0.0f;
  r.w = m.w ? vfc[(m.w - 1) * C] : 0.0f;

  // write-once stream -> bypass cache retention (global_store_b128 th:NT)
  __builtin_nontemporal_store(r, (vfloat4*)(out + o));
}

// ---------------------------------------------------------------------------
// Fallback path (workspace too small): classic zero + direct scatter.
// ---------------------------------------------------------------------------
__global__ void zero_out_kernel(float* __restrict__ out, int n4) {
  int i = blockIdx.x * blockDim.x + threadIdx.x;
  if (i < n4) {
    vfloat4 z = {0.f, 0.f, 0.f, 0.f};
    __builtin_nontemporal_store(z, (vfloat4*)(out + i * 4));
  }
}

__global__ void scatter_direct_kernel(const float* __restrict__ vf,
                                      const int4*  __restrict__ coords,
                                      float* __restrict__ out,
                                      int total, int C) {
  int t = blockIdx.x * blockDim.x + threadIdx.x;
  if (t >= total) return;
  int p = t / C;
  int c = t - p * C;
  int4 cd = coords[p];              // L2-cached, shared by C threads
  out[(cd.x * C + c) * NYNX + cd.z * NX_DIM + cd.w] = vf[t];
}

// ---------------------------------------------------------------------------
// Launch
// ---------------------------------------------------------------------------
extern "C" void kernel_launch(void* const* d_in, const int* in_sizes, int n_in,
                              void* d_out, int out_size, void* d_ws, size_t ws_size,
                              hipStream_t stream) {
  const float* vf     = (const float*)d_in[0];  // [N, C] f32
  const int4*  coords = (const int4*) d_in[1];  // [N, 4] i32
  float*       out    = (float*)d_out;          // [B, C, NY, NX] f32

  const int N = in_sizes[1] / 4;                // pillar count
  const int C = in_sizes[0] / N;                // channels (64)

  const int map_ints   = out_size / C;          // B * NY * NX
  const size_t map_req = (size_t)map_ints * sizeof(int);

  const int threads = 256;

  if (ws_size >= map_req) {
    // --- roofline path: map + single coalesced gather-write pass ---
    int* map = (int*)d_ws;

    int map4 = map_ints / 4;
    zero_map_kernel<<<(map4 + threads - 1) / threads, threads, 0, stream>>>(
        (int4*)map, map4);

    scatter_ids_kernel<<<(N + threads - 1) / threads, threads, 0, stream>>>(
        coords, map, N);

    int nfloat4 = out_size / 4;
    gather_out_kernel<<<(nfloat4 + threads - 1) / threads, threads, 0, stream>>>(
        vf, map, out, nfloat4, C);
  } else {
    // --- fallback: zero canvas, then direct scatter ---
    int n4 = out_size / 4;
    zero_out_kernel<<<(n4 + threads - 1) / threads, threads, 0, stream>>>(
        out, n4);

    int total = N * C;
    scatter_direct_kernel<<<(total + threads - 1) / threads, threads, 0, stream>>>(
        vf, coords, out, total, C);
  }
}